// QuanvolutionFilterQuantum_65481071405296
// MI455X (gfx1250) — compile-verified
//
#include <hip/hip_runtime.h>
#include <hip/hip_bf16.h>

// Quanvolution quantum filter, reduced to closed form.
//
// Per patch (theta0..theta3 = 2x2 stride-2 pixels):
//   d_w  = A_w*cos(theta_w) + B_w*sin(theta_w)
//   A_w  = cos(params[3w]) * cos(params[3w+1])   (Rx, Ry angles; Rz drops out)
//   B_w  = -sin(params[3w+1])
//   out  = d0*(W0 + d1*(W1 + d2*(W2 + d3*W3)))
//
// CDNA5 path: stage each 28x28 image into LDS with async global->LDS DMA
// (ASYNCcnt), split-wait, then compute from LDS.

#ifndef IMG_H
#define IMG_H 28
#endif
#define IMG_ELEMS (IMG_H * IMG_H)      // 784 floats
#define PATCHES_PER_IMG 196            // 14 x 14
#define QUADS_PER_IMG 196              // 784 / 4

__global__ void quanv_prep_kernel(const float* __restrict__ params,
                                  float* __restrict__ ab) {
    const int w = threadIdx.x;
    if (w < 4) {
        const float xi = params[3 * w + 0];  // Rx angle
        const float ph = params[3 * w + 1];  // Ry angle
        // Angles are O(1); native trig (v_cos_f32/v_sin_f32) is accurate here
        // and avoids the libm Payne-Hanek range-reduction blob.
        ab[w]     = __cosf(xi) * __cosf(ph); // A_w
        ab[4 + w] = -__sinf(ph);             // B_w
    }
}

__global__ __launch_bounds__(224) void quanv_main_kernel(
    const float* __restrict__ x,   // (B,1,28,28)
    const float* __restrict__ W,   // (1,4)
    const float* __restrict__ ab,  // A[4], B[4] from prep
    float* __restrict__ out) {     // (B,196)
    __shared__ float smem[IMG_ELEMS];

    const int t = threadIdx.x;
    const int b = blockIdx.x;

    // ---- Stage image b into LDS via CDNA5 async global->LDS DMA ----
    {
        const int q = (t < QUADS_PER_IMG) ? t : (QUADS_PER_IMG - 1); // clamp: dup write, same data
        unsigned lds_off = (unsigned)(unsigned long long)(const void*)&smem[q * 4];
        unsigned g_off   = ((unsigned)b * IMG_ELEMS + (unsigned)q * 4) * 4u; // byte offset
        asm volatile("global_load_async_to_lds_b128 %0, %1, %2"
                     :
                     : "v"(lds_off), "v"(g_off), "s"(x)
                     : "memory");
    }
    asm volatile("s_wait_asynccnt 0x0" ::: "memory");
    __syncthreads();

    if (t < PATCHES_PER_IMG) {
        const int i = t / 14;
        const int j = t % 14;
        const float* r = &smem[(i * 2) * IMG_H + j * 2];
        const float th0 = r[0];
        const float th1 = r[1];
        const float th2 = r[IMG_H + 0];
        const float th3 = r[IMG_H + 1];

        const float A0 = ab[0], A1 = ab[1], A2 = ab[2], A3 = ab[3];
        const float B0 = ab[4], B1 = ab[5], B2 = ab[6], B3 = ab[7];

        float s0, c0, s1, c1, s2, c2, s3, c3;
        __sincosf(th0, &s0, &c0);
        __sincosf(th1, &s1, &c1);
        __sincosf(th2, &s2, &c2);
        __sincosf(th3, &s3, &c3);

        const float d0 = fmaf(A0, c0, B0 * s0);
        const float d1 = fmaf(A1, c1, B1 * s1);
        const float d2 = fmaf(A2, c2, B2 * s2);
        const float d3 = fmaf(A3, c3, B3 * s3);

        const float w0 = W[0], w1 = W[1], w2 = W[2], w3 = W[3];
        // out = d0*(w0 + d1*(w1 + d2*(w2 + d3*w3)))
        const float res = d0 * fmaf(d1, fmaf(d2, fmaf(d3, w3, w2), w1), w0);

        out[b * PATCHES_PER_IMG + t] = res;
    }
}

extern "C" void kernel_launch(void* const* d_in, const int* in_sizes, int n_in,
                              void* d_out, int out_size, void* d_ws, size_t ws_size,
                              hipStream_t stream) {
    const float* x      = (const float*)d_in[0];  // (B,1,28,28)
    const float* params = (const float*)d_in[1];  // (12,)
    const float* W      = (const float*)d_in[2];  // (1,4)
    float* out          = (float*)d_out;          // (B,196)
    float* ab           = (float*)d_ws;           // A[4], B[4]

    const int nb = in_sizes[0] / IMG_ELEMS;       // batch (8192)

    quanv_prep_kernel<<<1, 32, 0, stream>>>(params, ab);
    quanv_main_kernel<<<nb, 224, 0, stream>>>(x, W, ab, out);
}